// StimulusModel_90426241450613
// MI455X (gfx1250) — compile-verified
//
#include <hip/hip_runtime.h>

// Elementwise scan, store-bandwidth bound (2 GiB of output at 23.3 TB/s ~ 90us floor).
//   g  = 2*sigmoid(u - ut)           = tanh((u-ut)/2) + 1
//   u' = clip(u + 0.05 - 0.1*g, ...) = med3(fma(-0.1, tanh, u - 0.05), 0.01, 0.2)
// One v_tanh_f32 per element-step; vec4 non-temporal b128 stores, 32-bit
// running row offset so stores stay in saddr+voffset scale_offset form.

typedef float v4f __attribute__((ext_vector_type(4)));

__global__ __launch_bounds__(256) void StimulusModel_scan_kernel(
    const float* __restrict__ u_init,
    const float* __restrict__ u_trig_ptr,
    float* __restrict__ out,
    int nvec,      // B / 4
    int steps)
{
    const unsigned gid = blockIdx.x * blockDim.x + threadIdx.x;
    if ((int)gid >= nvec) return;

    const float ut2 = 0.5f * u_trig_ptr[0];         // uniform scalar

    v4f u = ((const v4f* __restrict__)u_init)[gid];

    v4f* __restrict__ out_u = (v4f*)out;
    v4f* __restrict__ out_g = out_u + (size_t)steps * (size_t)nvec;

    unsigned off = gid;                              // max steps*nvec = 2^26, fits u32

    #pragma unroll 2
    for (int t = 0; t < steps; ++t) {
        v4f g, un;
        #pragma unroll
        for (int k = 0; k < 4; ++k) {
            float z2 = __builtin_fmaf(0.5f, u[k], -ut2);     // (u - ut)/2
            float th = __builtin_amdgcn_tanhf(z2);           // v_tanh_f32
            float gg = th + 1.0f;                            // g = tanh+1 in [0,2]
            float v  = __builtin_fmaf(-0.1f, th, u[k] - 0.05f); // u +0.05 -0.1*g
            v = __builtin_amdgcn_fmed3f(v, 0.01f, 0.2f);     // clip via v_med3
            g[k]  = gg;
            un[k] = v;
        }
        u = un;

        __builtin_nontemporal_store(un, out_u + off);        // global_store_b128 th:NT
        __builtin_nontemporal_store(g,  out_g + off);        // global_store_b128 th:NT
        off += (unsigned)nvec;
    }
}

extern "C" void kernel_launch(void* const* d_in, const int* in_sizes, int n_in,
                              void* d_out, int out_size, void* d_ws, size_t ws_size,
                              hipStream_t stream) {
    const float* u_initial = (const float*)d_in[0];
    const float* u_trigger = (const float*)d_in[1];
    // steps is derivable on host: out_size = 2 * steps * B
    const int B     = in_sizes[0];
    const int steps = (B > 0) ? (out_size / (2 * B)) : 0;
    const int nvec  = B >> 2;                        // B = 2^20, divisible by 4

    float* out = (float*)d_out;

    const int block = 256;
    const int grid  = (nvec + block - 1) / block;
    StimulusModel_scan_kernel<<<grid, block, 0, stream>>>(u_initial, u_trigger,
                                                          out, nvec, steps);
}